// GraphAntiSymmetricNN_GraphProp_4406636446398
// MI455X (gfx1250) — compile-verified
//
#include <hip/hip_runtime.h>
#include <math.h>

#define HD 128
#define GAMMA_C 0.1f
#define EPS_C 0.1f
#define NEG_SLOPE_C 0.01f
#define NUM_ITERS_C 5
#define LDSW 130   // padded LDS row stride (floats): 520B, 8B-aligned, bank-conflict free

typedef __attribute__((ext_vector_type(2))) float v2f;
typedef __attribute__((ext_vector_type(8))) float v8f;

// ---- A = W - W^T - gamma*I ----------------------------------------------
__global__ void prep_A_kernel(const float* __restrict__ W, float* __restrict__ A) {
  int i = blockIdx.x * blockDim.x + threadIdx.x;
  if (i < HD * HD) {
    int r = i >> 7, c = i & 127;
    float v = W[r * HD + c] - W[c * HD + r];
    if (r == c) v -= GAMMA_C;
    A[i] = v;
  }
}

__global__ void copy4_kernel(const float* __restrict__ in, float* __restrict__ out, int n4) {
  int i = blockIdx.x * blockDim.x + threadIdx.x;
  if (i < n4) ((float4*)out)[i] = ((const float4*)in)[i];
}

__global__ void zero4_kernel(float* __restrict__ out, int n4) {
  int i = blockIdx.x * blockDim.x + threadIdx.x;
  if (i < n4) ((float4*)out)[i] = make_float4(0.f, 0.f, 0.f, 0.f);
}

// ---- Y = X @ Wm^T  (mode 0)  |  Y = X + eps*tanh(X@Wm^T + agg + bias) (mode 1)
// Block: 256 threads = 8 waves; each wave owns a 16-row strip; block covers 128 rows.
__global__ __launch_bounds__(256)
void gemm_wmma_kernel(const float* __restrict__ X, const float* __restrict__ Wm,
                      const float* __restrict__ agg, const float* __restrict__ bias,
                      float* __restrict__ Y, int n, int mode)
{
  __shared__ float lw[HD * LDSW];
  int tid = threadIdx.x;
  for (int i = tid; i < HD * HD; i += 256) {
    int r = i >> 7, c = i & 127;
    lw[r * LDSW + c] = Wm[i];
  }
  __syncthreads();

  int wave  = tid >> 5;
  int lane  = tid & 31;
  int row16 = lane & 15;      // row/col within 16-tile
  int khalf = lane >> 4;      // selects K pair {0,1} vs {2,3}
  int row0  = blockIdx.x * 128 + wave * 16;

  int arow = row0 + row16;
  if (arow > n - 1) arow = n - 1;      // clamp loads for tail block (stores guarded)
  const float* xrow = X + (long long)arow * HD;

  v8f acc[8];
#pragma unroll
  for (int i = 0; i < 8; ++i) acc[i] = (v8f){0.f,0.f,0.f,0.f,0.f,0.f,0.f,0.f};

#pragma unroll 4
  for (int k = 0; k < 32; ++k) {            // K = 128 in steps of 4
    v2f a = *(const v2f*)(xrow + k * 4 + khalf * 2);
    const float* lb = &lw[row16 * LDSW + k * 4 + khalf * 2];
#pragma unroll
    for (int nt = 0; nt < 8; ++nt) {        // N = 128 in 16-col tiles
      v2f b = *(const v2f*)(lb + nt * 16 * LDSW);
      acc[nt] = __builtin_amdgcn_wmma_f32_16x16x4_f32(
          false, a, false, b, (short)0, acc[nt], false, false);
    }
  }

  int mbase = row0 + khalf * 8;             // D layout: M = r + 8*(lane/16)
#pragma unroll
  for (int nt = 0; nt < 8; ++nt) {
    int col = nt * 16 + row16;              // N = lane%16
#pragma unroll
    for (int r = 0; r < 8; ++r) {
      int orow = mbase + r;
      if (orow < n) {
        long long idx = (long long)orow * HD + col;
        float v = acc[nt][r];
        if (mode == 0) {
          Y[idx] = v;
        } else {
          float t = v + agg[idx] + bias[col];
          Y[idx] = X[idx] + EPS_C * tanhf(t);
        }
      }
    }
  }
}

// ---- agg[dst] += neigh[src] over edges (L2-resident atomics) -------------
__global__ void scatter_edges_kernel(const int* __restrict__ src, const int* __restrict__ dst,
                                     const float* __restrict__ neigh, float* __restrict__ agg,
                                     int E)
{
  long long gid = (long long)blockIdx.x * blockDim.x + threadIdx.x;
  int e = (int)(gid >> 7);
  int j = (int)(gid & 127);
  if (e < E) {
    int s = src[e], d = dst[e];
    atomicAdd(&agg[(long long)d * HD + j], neigh[(long long)s * HD + j]);
  }
}

// ---- pooling (add/max/mean via sorted-batch binary search) + 2-layer MLP --
__device__ __forceinline__ int lower_bound_i(const int* __restrict__ a, int n, int key) {
  int lo = 0, hi = n;
  while (lo < hi) { int mid = (lo + hi) >> 1; if (a[mid] < key) lo = mid + 1; else hi = mid; }
  return lo;
}

__global__ __launch_bounds__(192)
void pool_mlp_kernel(const float* __restrict__ x, const int* __restrict__ batch, int n,
                     const float* __restrict__ l1w, const float* __restrict__ l1b,
                     const float* __restrict__ l2w, const float* __restrict__ l2b,
                     float* __restrict__ out)
{
  __shared__ float pooled[3 * HD];
  __shared__ float hbuf[192];
  int g = blockIdx.x, t = threadIdx.x;
  int lo = lower_bound_i(batch, n, g);
  int hi = lower_bound_i(batch, n, g + 1);
  if (t < HD) {
    float s = 0.f, m = -3.402823466e38f;
    for (int i = lo; i < hi; ++i) {
      float v = x[(long long)i * HD + t];
      s += v;
      m = fmaxf(m, v);
    }
    int cnt = hi - lo;
    pooled[t]          = s;
    pooled[HD + t]     = (cnt > 0) ? m : 0.f;
    pooled[2 * HD + t] = s / fmaxf((float)cnt, 1.f);
  }
  __syncthreads();
  {
    float a1 = l1b[t];
    const float* wrow = l1w + (long long)t * (3 * HD);
    for (int k = 0; k < 3 * HD; ++k) a1 += pooled[k] * wrow[k];
    hbuf[t] = (a1 > 0.f) ? a1 : NEG_SLOPE_C * a1;
  }
  __syncthreads();
  if (t < 64) {
    float a2 = l2b[t];
    const float* w2 = l2w + t * 192;
    for (int k = 0; k < 192; ++k) a2 += hbuf[k] * w2[k];
    out[g * 64 + t] = (a2 > 0.f) ? a2 : NEG_SLOPE_C * a2;
  }
}

extern "C" void kernel_launch(void* const* d_in, const int* in_sizes, int n_in,
                              void* d_out, int out_size, void* d_ws, size_t ws_size,
                              hipStream_t stream) {
  const float* x_in  = (const float*)d_in[0];
  const int*   ei    = (const int*)d_in[1];   // [2,E] : src row then dst row
  const int*   batch = (const int*)d_in[2];
  const float* W     = (const float*)d_in[3];
  const float* bias  = (const float*)d_in[4];
  const float* lin_w = (const float*)d_in[5];
  const float* l1w   = (const float*)d_in[6];
  const float* l1b   = (const float*)d_in[7];
  const float* l2w   = (const float*)d_in[8];
  const float* l2b   = (const float*)d_in[9];
  float* out = (float*)d_out;

  int N = in_sizes[0] / HD;
  int E = in_sizes[1] / 2;
  int G = out_size / 64;

  float* ws   = (float*)d_ws;
  float* Amat = ws;                                   // 128*128
  float* buf0 = Amat + HD * HD;                       // N*128
  float* buf1 = buf0 + (size_t)N * HD;                // N*128 (neigh / x_next)
  float* agg  = buf1 + (size_t)N * HD;                // N*128

  int n4 = (N * HD) / 4;
  prep_A_kernel<<<(HD * HD + 255) / 256, 256, 0, stream>>>(W, Amat);
  copy4_kernel<<<(n4 + 255) / 256, 256, 0, stream>>>(x_in, buf0, n4);

  float* xcur = buf0;
  float* tmp  = buf1;
  int gemmBlocks = (N + 127) / 128;
  long long sthreads = (long long)E * HD;
  int sblocks = (int)((sthreads + 255) / 256);

  for (int it = 0; it < NUM_ITERS_C; ++it) {
    // neigh = x @ lin_w^T  (into tmp)
    gemm_wmma_kernel<<<gemmBlocks, 256, 0, stream>>>(xcur, lin_w, nullptr, nullptr, tmp, N, 0);
    zero4_kernel<<<(n4 + 255) / 256, 256, 0, stream>>>(agg, n4);
    scatter_edges_kernel<<<sblocks, 256, 0, stream>>>(ei, ei + E, tmp, agg, E);
    // x_next = x + eps*tanh(x@A^T + agg + bias)  (into tmp, neigh is dead now)
    gemm_wmma_kernel<<<gemmBlocks, 256, 0, stream>>>(xcur, Amat, agg, bias, tmp, N, 1);
    float* sw = xcur; xcur = tmp; tmp = sw;
  }

  pool_mlp_kernel<<<G, 192, 0, stream>>>(xcur, batch, N, l1w, l1b, l2w, l2b, out);
}